// TopKFilter_7567732376177
// MI455X (gfx1250) — compile-verified
//
#include <hip/hip_runtime.h>
#include <stdint.h>

// Top-k row filter: keep k largest per row, zero the rest.
// One workgroup (1024 threads = 32 wave32 waves) per row.
// Row (24576 fp32 = 96 KB) staged once in CDNA5's 320KB LDS via
// GLOBAL_LOAD_ASYNC_TO_LDS_B128; radix select runs entirely out of LDS
// -> exactly 1 global read + 1 non-temporal global write per row
// (~66us HBM floor at 23.3 TB/s).

constexpr int ROW  = 24576;
constexpr int TPB  = 1024;
constexpr int EPT  = ROW / TPB;       // 24 elements / thread
constexpr int V4PT = (ROW / 4) / TPB; // 6 uint4 / thread

typedef __attribute__((address_space(3))) uint8_t lds_byte_t;
typedef float __attribute__((ext_vector_type(4))) fvec4;   // native vector for NT store

// Order-preserving float->uint key: larger float <=> larger key.
__device__ __forceinline__ uint32_t f2key(uint32_t f) {
    return f ^ (uint32_t(int32_t(f) >> 31) | 0x80000000u);
}

__global__ __launch_bounds__(TPB, 1)
void topk_filter_kernel(const float* __restrict__ x,
                        float* __restrict__ out,
                        const int* __restrict__ kptr) {
    __shared__ uint32_t     rowBits[ROW];   // 96 KB: the whole row
    __shared__ unsigned int hist[256];      // radix histogram
    __shared__ uint32_t     scanArr[TPB];   // per-chunk equal-count base
    __shared__ uint32_t     waveSum[32];    // per-wave scan totals
    __shared__ uint32_t     s_bin;
    __shared__ uint32_t     s_rk;

    const int    t      = (int)threadIdx.x;
    const int    lane   = t & 31;
    const int    wave   = t >> 5;
    const size_t rowOff = (size_t)blockIdx.x * (size_t)ROW;

    const float* __restrict__ srcBase = x + rowOff;
    fvec4*       __restrict__ dst     = (fvec4*)(out + rowOff);
    uint4* rowV4 = (uint4*)rowBits;

    // ---- Stage 0: async DMA fill global -> LDS (b128 per lane, ASYNCcnt) ----
    {
        const uint32_t ldsBase = (uint32_t)(uintptr_t)(lds_byte_t*)rowBits;
        const uint32_t lOff = ldsBase + 16u * (uint32_t)t;  // VDST: LDS byte addr
        const uint32_t gOff = 16u * (uint32_t)t;            // VADDR: global byte off
        // INST_OFFSET is added to BOTH the LDS and global address (ISA 08 §4.4),
        // and our LDS layout mirrors the global row 1:1, so shared offsets work.
        asm volatile("global_load_async_to_lds_b128 %0, %1, %2 offset:0"
                     :: "v"(lOff), "v"(gOff), "s"(srcBase) : "memory");
        asm volatile("global_load_async_to_lds_b128 %0, %1, %2 offset:16384"
                     :: "v"(lOff), "v"(gOff), "s"(srcBase) : "memory");
        asm volatile("global_load_async_to_lds_b128 %0, %1, %2 offset:32768"
                     :: "v"(lOff), "v"(gOff), "s"(srcBase) : "memory");
        asm volatile("global_load_async_to_lds_b128 %0, %1, %2 offset:49152"
                     :: "v"(lOff), "v"(gOff), "s"(srcBase) : "memory");
        asm volatile("global_load_async_to_lds_b128 %0, %1, %2 offset:65536"
                     :: "v"(lOff), "v"(gOff), "s"(srcBase) : "memory");
        asm volatile("global_load_async_to_lds_b128 %0, %1, %2 offset:81920"
                     :: "v"(lOff), "v"(gOff), "s"(srcBase) : "memory");
        asm volatile("s_wait_asynccnt 0x0" ::: "memory");
    }

    uint32_t rk     = (uint32_t)kptr[0];  // elements still to place
    uint32_t prefix = 0;                  // high bits of k-th largest key
    __syncthreads();

    // ---- Stage 1: 4-pass MSB radix select (256 bins/pass) ----
    for (int pass = 0; pass < 4; ++pass) {
        const int shift = 24 - pass * 8;

        if (t < 256) hist[t] = 0u;
        __syncthreads();

        for (int j = 0; j < EPT; ++j) {
            uint32_t key = f2key(rowBits[t + j * TPB]);
            if (pass == 0 || (key >> (shift + 8)) == prefix)
                atomicAdd(&hist[(key >> shift) & 0xFFu], 1u);
        }
        __syncthreads();

        // Wave 0 only: suffix-sum 256 bins in registers (8 bins/lane, wave32
        // shuffles), pick the unique bin with excl < rk <= incl.
        if (t < 32) {
            uint32_t c[8], s[8];
            #pragma unroll
            for (int j = 0; j < 8; ++j) c[j] = hist[lane * 8 + j];
            uint32_t run = 0;
            #pragma unroll
            for (int j = 7; j >= 0; --j) { run += c[j]; s[j] = run; }
            uint32_t acc = run;                       // lane total
            #pragma unroll
            for (int off = 1; off < 32; off <<= 1) {  // inclusive suffix scan
                uint32_t o = __shfl_down(acc, off, 32);
                if (lane + off < 32) acc += o;
            }
            const uint32_t crossExcl = acc - run;     // sum of lanes > lane
            #pragma unroll
            for (int j = 0; j < 8; ++j) {
                uint32_t Sincl = crossExcl + s[j];
                uint32_t Sexcl = Sincl - c[j];
                if (Sexcl < rk && Sincl >= rk) {
                    s_bin = (uint32_t)(lane * 8 + j);
                    s_rk  = rk - Sexcl;
                }
            }
        }
        __syncthreads();
        prefix = (prefix << 8) | s_bin;
        rk     = s_rk;
    }

    const uint32_t thKey = prefix;   // exact key of k-th largest element
    // rk = how many elements EQUAL to thKey to keep (earliest indices first)

    // ---- Stage 2a: index-ordered ranks for threshold-equal elements ----
    // Chunk t = contiguous [t*EPT, (t+1)*EPT); chunk order == index order.
    uint32_t eq = 0;
    {
        const int base = t * EPT;
        #pragma unroll
        for (int j = 0; j < EPT; ++j)
            eq += (f2key(rowBits[base + j]) == thKey) ? 1u : 0u;
    }
    // Two-level exclusive scan over 1024 chunk counts:
    uint32_t val = eq;
    #pragma unroll
    for (int off = 1; off < 32; off <<= 1) {          // wave inclusive scan
        uint32_t o = __shfl_up(val, off, 32);
        if (lane >= off) val += o;
    }
    if (lane == 31) waveSum[wave] = val;              // wave totals
    __syncthreads();
    if (t < 32) {                                     // scan 32 wave totals
        uint32_t wv = waveSum[t];
        uint32_t sc = wv;
        #pragma unroll
        for (int off = 1; off < 32; off <<= 1) {
            uint32_t o = __shfl_up(sc, off, 32);
            if (t >= off) sc += o;
        }
        waveSum[t] = sc - wv;                         // exclusive wave base
    }
    __syncthreads();
    scanArr[t] = waveSum[wave] + (val - eq);          // exclusive chunk base
    __syncthreads();

    // ---- Stage 2b: masked output (coalesced b128, non-temporal stores) ----
    #pragma unroll
    for (int j = 0; j < V4PT; ++j) {
        const int vidx = t + j * TPB;
        uint4 b = rowV4[vidx];
        uint32_t bw[4] = { b.x, b.y, b.z, b.w };
        fvec4 o;
        #pragma unroll
        for (int e = 0; e < 4; ++e) {
            const uint32_t key = f2key(bw[e]);
            float r = 0.0f;
            if (key > thKey) {
                r = __uint_as_float(bw[e]);
            } else if (key == thKey) {           // rare path (ties)
                const int i = 4 * vidx + e;
                const int c = i / EPT;
                uint32_t rank = scanArr[c];
                for (int m = c * EPT; m < i; ++m)
                    rank += (f2key(rowBits[m]) == thKey) ? 1u : 0u;
                if (rank < rk) r = __uint_as_float(bw[e]);
            }
            o[e] = r;
        }
        // Write-once output stream: bypass cache retention (th:TH_STORE_NT).
        __builtin_nontemporal_store(o, &dst[vidx]);
    }
}

extern "C" void kernel_launch(void* const* d_in, const int* in_sizes, int n_in,
                              void* d_out, int out_size, void* d_ws, size_t ws_size,
                              hipStream_t stream) {
    const float* x  = (const float*)d_in[0];
    const int*   kp = (const int*)d_in[1];   // k as 1-element device int array
    float*       o  = (float*)d_out;

    const int nrows = in_sizes[0] / ROW;     // 8192
    topk_filter_kernel<<<nrows, TPB, 0, stream>>>(x, o, kp);
}